// RelationalLayer_10565619548782
// MI455X (gfx1250) — compile-verified
//
#include <hip/hip_runtime.h>

typedef float v2f __attribute__((ext_vector_type(2)));
typedef float v8f __attribute__((ext_vector_type(8)));

#define BATCH 8
#define NROW 256
#define DIM 128

// One workgroup = one (batch b, 16-row i-tile).
// 256 threads = 8 waves. LDS: c_s[256][128] (128KB) + a_s[16][128] (8KB).
// Phase 1: WMMA f32 16x16x4 GEMMs  c_s = X_b @ W2 (full 256x128),
//                                  a_s = X_b[i0:i0+16,:] @ W1.
// Phase 2: out[b, i0+i, d] = sum_j relu(a_s[i][d] + bias[d] + c_s[j][d]).
__global__ __launch_bounds__(256, 2)
void relational_fused_kernel(const float* __restrict__ x,
                             const float* __restrict__ W,
                             const float* __restrict__ bias,
                             float* __restrict__ out) {
  extern __shared__ float smem[];
  float* c_s = smem;                  // [256][128]
  float* a_s = smem + NROW * DIM;     // [16][128]

  const int blk   = blockIdx.x;
  const int b     = blk >> 4;         // batch index
  const int itile = blk & 15;         // which 16-row tile of i
  const int tid   = threadIdx.x;
  const int wave  = tid >> 5;         // 0..7
  const int lane  = tid & 31;
  const int lrow  = lane & 15;        // M (A) or N (B) index within tile
  const int g     = lane >> 4;        // K-half selector (0 or 1)

  const float* Xb = x + b * (NROW * DIM);
  const float* W1 = W;                // W1[k][n], row-major 128x128
  const float* W2 = W + DIM * DIM;    // W2[k][n] = W[128+k][n]

  // ---------- GEMM: c_s = Xb (256x128) @ W2 (128x128) ----------
  // 16 m-tiles x 8 n-tiles = 128 tiles of 16x16; wave w takes t = w, w+8, ...
  for (int t = wave; t < 128; t += 8) {
    const int mt = t >> 3;            // 0..15
    const int nt = t & 7;             // 0..7
    v8f acc = {};
    const float* arow = Xb + (mt * 16 + lrow) * DIM;   // A: row of X
    const float* bcol = W2 + (nt * 16 + lrow);         // B: column of W2
    #pragma unroll 4
    for (int k0 = 0; k0 < DIM; k0 += 4) {
      const int ka = k0 + 2 * g;
      v2f afrag = { arow[ka], arow[ka + 1] };
      v2f bfrag = { bcol[ka * DIM], bcol[(ka + 1) * DIM] };
      acc = __builtin_amdgcn_wmma_f32_16x16x4_f32(
          /*neg_a=*/false, afrag, /*neg_b=*/false, bfrag,
          /*c_mod=*/(short)0, acc, /*reuse_a=*/false, /*reuse_b=*/false);
    }
    #pragma unroll
    for (int r = 0; r < 8; ++r) {
      // D layout: VGPR r, lane-group g -> M = r + 8g, N = lrow
      c_s[(mt * 16 + r + 8 * g) * DIM + nt * 16 + lrow] = acc[r];
    }
  }

  // ---------- GEMM: a_s = Xb[i0:i0+16, :] @ W1 ; wave w does n-tile w ----------
  {
    const int i0 = itile * 16;
    v8f acc = {};
    const float* arow = Xb + (i0 + lrow) * DIM;
    const float* bcol = W1 + (wave * 16 + lrow);
    #pragma unroll 4
    for (int k0 = 0; k0 < DIM; k0 += 4) {
      const int ka = k0 + 2 * g;
      v2f afrag = { arow[ka], arow[ka + 1] };
      v2f bfrag = { bcol[ka * DIM], bcol[(ka + 1) * DIM] };
      acc = __builtin_amdgcn_wmma_f32_16x16x4_f32(
          false, afrag, false, bfrag, (short)0, acc, false, false);
    }
    #pragma unroll
    for (int r = 0; r < 8; ++r) {
      a_s[(r + 8 * g) * DIM + wave * 16 + lrow] = acc[r];
    }
  }

  __syncthreads();

  // ---------- Reduction: each thread owns (d, 8 consecutive i) ----------
  const int d    = tid & (DIM - 1);
  const int half = tid >> 7;          // 0 or 1 -> i offset 0 or 8
  const float bd = bias[d];
  float s[8], acc[8];
  #pragma unroll
  for (int k = 0; k < 8; ++k) {
    s[k]   = a_s[(half * 8 + k) * DIM + d] + bd;
    acc[k] = 0.0f;
  }
  #pragma unroll 4
  for (int j = 0; j < NROW; ++j) {
    const float v = c_s[j * DIM + d];
    #pragma unroll
    for (int k = 0; k < 8; ++k) {
      acc[k] += fmaxf(s[k] + v, 0.0f);
    }
  }
  float* op = out + ((b * NROW) + itile * 16 + half * 8) * DIM + d;
  #pragma unroll
  for (int k = 0; k < 8; ++k) {
    op[k * DIM] = acc[k];
  }
}

extern "C" void kernel_launch(void* const* d_in, const int* in_sizes, int n_in,
                              void* d_out, int out_size, void* d_ws, size_t ws_size,
                              hipStream_t stream) {
  (void)in_sizes; (void)n_in; (void)d_ws; (void)ws_size; (void)out_size;
  const float* x    = (const float*)d_in[0];   // (8,256,128)
  const float* W    = (const float*)d_in[1];   // (256,128)
  const float* bias = (const float*)d_in[2];   // (128,)
  float* out        = (float*)d_out;           // (8,256,128)

  const size_t shmem = (size_t)(NROW * DIM + 16 * DIM) * sizeof(float); // 136 KB
  (void)hipFuncSetAttribute((const void*)relational_fused_kernel,
                            hipFuncAttributeMaxDynamicSharedMemorySize,
                            (int)shmem);

  dim3 grid(BATCH * (NROW / 16));   // 128 workgroups
  dim3 block(256);                  // 8 wave32 waves
  relational_fused_kernel<<<grid, block, shmem, stream>>>(x, W, bias, out);
}